// VQ_49830210568543
// MI455X (gfx1250) — compile-verified
//
#include <hip/hip_runtime.h>

#define DIM 512
#define KCODES 2048
#define ROWS_PER_BLOCK 16
#define BT (16 * 1024)                 // 16384 vectors
#define NBLOCKS (BT / ROWS_PER_BLOCK)  // 1024 workgroups

typedef __attribute__((ext_vector_type(2))) float v2f;
typedef __attribute__((ext_vector_type(8))) float v8f;

// ---------------------------------------------------------------------------
// Main kernel: one workgroup = one 16-row x tile; 8 waves split the 2048
// codebook entries (128 N-tiles of 16). Distances via fp32 WMMA 16x16x4.
// ---------------------------------------------------------------------------
__global__ __launch_bounds__(256) void vq_main_kernel(
    const float* __restrict__ x,        // (16384, 512)
    const float* __restrict__ cb,       // (2048, 512)
    float* __restrict__ out_q,          // (16384, 512)
    float* __restrict__ out_idx,        // (16384,) indices as float
    float* __restrict__ partials)       // (NBLOCKS,) commit-loss partials
{
    __shared__ float xlds[ROWS_PER_BLOCK * DIM];   // 32 KB x tile
    __shared__ float red_d[8][ROWS_PER_BLOCK];
    __shared__ int   red_i[8][ROWS_PER_BLOCK];
    __shared__ int   fidx[ROWS_PER_BLOCK];
    __shared__ float rsum[256];

    const int tid   = threadIdx.x;
    const int wave  = tid >> 5;
    const int lane  = tid & 31;
    const int nl    = lane & 15;   // column-in-tile / A row
    const int hi    = lane >> 4;   // half-wave selector (K pair select)
    const int block = blockIdx.x;

    // ---- stage the 16x512 x tile into LDS (float4, fully coalesced) ----
    {
        const float4* src = (const float4*)(x + (size_t)block * ROWS_PER_BLOCK * DIM);
        float4* dst = (float4*)xlds;
        #pragma unroll
        for (int v = tid; v < ROWS_PER_BLOCK * DIM / 4; v += 256) dst[v] = src[v];
    }
    __syncthreads();

    // running per-lane argmin state: acc VGPR j holds row (j + 8*hi), col nl
    float bestd[8];
    int   besti[8];
    #pragma unroll
    for (int j = 0; j < 8; ++j) { bestd[j] = 3.0e38f; besti[j] = 0; }

    // A fragment source: row = lane%16, K pair {k0+2*hi, k0+1+2*hi}
    const float* aptr = xlds + nl * DIM + 2 * hi;

    // each wave owns N-tiles [wave*16, wave*16+16), processed 4 at a time
    for (int g = 0; g < 4; ++g) {
        const int tile0 = wave * 16 + g * 4;
        v8f a0 = {}, a1 = {}, a2 = {}, a3 = {};
        float cs0 = 0.f, cs1 = 0.f, cs2 = 0.f, cs3 = 0.f;
        const float* b0 = cb + ((size_t)(tile0 + 0) * 16 + nl) * DIM + 2 * hi;
        const float* b1 = b0 + 16 * DIM;
        const float* b2 = b1 + 16 * DIM;
        const float* b3 = b2 + 16 * DIM;

        #pragma unroll 4
        for (int k0 = 0; k0 < DIM; k0 += 4) {
            v2f av  = *(const v2f*)(aptr + k0);  // ds_load_b64
            v2f bv0 = *(const v2f*)(b0 + k0);    // global_load_b64 (L2-resident)
            v2f bv1 = *(const v2f*)(b1 + k0);
            v2f bv2 = *(const v2f*)(b2 + k0);
            v2f bv3 = *(const v2f*)(b3 + k0);
            // 4 independent fp32 WMMA accumulation chains
            a0 = __builtin_amdgcn_wmma_f32_16x16x4_f32(false, av, false, bv0, (short)0, a0, false, false);
            a1 = __builtin_amdgcn_wmma_f32_16x16x4_f32(false, av, false, bv1, (short)0, a1, false, false);
            a2 = __builtin_amdgcn_wmma_f32_16x16x4_f32(false, av, false, bv2, (short)0, a2, false, false);
            a3 = __builtin_amdgcn_wmma_f32_16x16x4_f32(false, av, false, bv3, (short)0, a3, false, false);
            // ||c||^2 partials ride along on the VALU (co-executes with XDL)
            cs0 += bv0.x * bv0.x + bv0.y * bv0.y;
            cs1 += bv1.x * bv1.x + bv1.y * bv1.y;
            cs2 += bv2.x * bv2.x + bv2.y * bv2.y;
            cs3 += bv3.x * bv3.x + bv3.y * bv3.y;
        }
        // each lane covered half the K range; partner lane (xor 16) has the rest
        cs0 += __shfl_xor(cs0, 16, 32);
        cs1 += __shfl_xor(cs1, 16, 32);
        cs2 += __shfl_xor(cs2, 16, 32);
        cs3 += __shfl_xor(cs3, 16, 32);

        // surrogate distance: ||c||^2 - 2*x.c  (||x||^2 constant per row)
        #pragma unroll
        for (int j = 0; j < 8; ++j) {
            float d; int i;
            d = cs0 - 2.0f * a0[j]; i = (tile0 + 0) * 16 + nl;
            if (d < bestd[j] || (d == bestd[j] && i < besti[j])) { bestd[j] = d; besti[j] = i; }
            d = cs1 - 2.0f * a1[j]; i = (tile0 + 1) * 16 + nl;
            if (d < bestd[j] || (d == bestd[j] && i < besti[j])) { bestd[j] = d; besti[j] = i; }
            d = cs2 - 2.0f * a2[j]; i = (tile0 + 2) * 16 + nl;
            if (d < bestd[j] || (d == bestd[j] && i < besti[j])) { bestd[j] = d; besti[j] = i; }
            d = cs3 - 2.0f * a3[j]; i = (tile0 + 3) * 16 + nl;
            if (d < bestd[j] || (d == bestd[j] && i < besti[j])) { bestd[j] = d; besti[j] = i; }
        }
    }

    // ---- butterfly argmin across the 16 lanes of each half-wave ----
    #pragma unroll
    for (int j = 0; j < 8; ++j) {
        float d = bestd[j]; int i = besti[j];
        #pragma unroll
        for (int m = 1; m < 16; m <<= 1) {
            float od = __shfl_xor(d, m, 32);
            int   oi = __shfl_xor(i, m, 32);
            if (od < d || (od == d && oi < i)) { d = od; i = oi; }
        }
        bestd[j] = d; besti[j] = i;
    }
    if (nl == 0) {  // lanes 0 and 16 publish rows [0..7] and [8..15]
        #pragma unroll
        for (int j = 0; j < 8; ++j) {
            red_d[wave][j + 8 * hi] = bestd[j];
            red_i[wave][j + 8 * hi] = besti[j];
        }
    }
    __syncthreads();

    // ---- combine across the 8 waves (waves cover ascending index ranges) ----
    if (tid < ROWS_PER_BLOCK) {
        const int r = tid;
        float d = red_d[0][r]; int i = red_i[0][r];
        #pragma unroll
        for (int w = 1; w < 8; ++w) {
            float od = red_d[w][r]; int oi = red_i[w][r];
            if (od < d || (od == d && oi < i)) { d = od; i = oi; }
        }
        fidx[r] = i;
        out_idx[block * ROWS_PER_BLOCK + r] = (float)i;
    }
    __syncthreads();

    // ---- gather quantized rows, emit, accumulate commitment-loss partial ----
    float partial = 0.f;
    for (int e = tid; e < ROWS_PER_BLOCK * DIM; e += 256) {
        const int r  = e >> 9;        // /512
        const int dd = e & (DIM - 1);
        const float q  = cb[(size_t)fidx[r] * DIM + dd];
        const float xv = xlds[e];
        out_q[(size_t)(block * ROWS_PER_BLOCK + r) * DIM + dd] = q;
        const float df = q - xv;
        partial += df * df;
    }
    rsum[tid] = partial;
    __syncthreads();
    #pragma unroll
    for (int s = 128; s > 0; s >>= 1) {
        if (tid < s) rsum[tid] += rsum[tid + s];
        __syncthreads();
    }
    if (tid == 0) partials[block] = rsum[0];
}

// ---------------------------------------------------------------------------
// Deterministic loss finalize: fixed-order sum of per-block partials.
// ---------------------------------------------------------------------------
__global__ void vq_loss_finalize_kernel(const float* __restrict__ partials,
                                        float* __restrict__ out_loss)
{
    if (threadIdx.x == 0 && blockIdx.x == 0) {
        float s = 0.f;
        for (int i = 0; i < NBLOCKS; ++i) s += partials[i];
        *out_loss = s / (float)((size_t)BT * DIM);
    }
}

extern "C" void kernel_launch(void* const* d_in, const int* in_sizes, int n_in,
                              void* d_out, int out_size, void* d_ws, size_t ws_size,
                              hipStream_t stream)
{
    const float* x  = (const float*)d_in[0];   // (16,1024,512) f32
    const float* cb = (const float*)d_in[1];   // (2048,512) f32

    float* out   = (float*)d_out;
    float* out_q   = out;                        // 16384*512 quantized
    float* out_idx = out + (size_t)BT * DIM;     // 16384 indices (as float)
    float* out_ls  = out_idx + BT;               // 1 scalar loss

    float* partials = (float*)d_ws;              // NBLOCKS floats of scratch

    vq_main_kernel<<<NBLOCKS, 256, 0, stream>>>(x, cb, out_q, out_idx, partials);
    vq_loss_finalize_kernel<<<1, 32, 0, stream>>>(partials, out_ls);
}